// CSR_10359461118634
// MI455X (gfx1250) — compile-verified
//
#include <hip/hip_runtime.h>
#include <math.h>

// Problem constants (match reference)
#define B_SZ 2048
#define D_SZ 512
#define L_SZ 32768
#define K_TOP 32
#define K_AUX 256
#define DEAD_THRESHOLD 100
#define EPS_LN 1e-5f

typedef __attribute__((ext_vector_type(2))) float v2f;
typedef __attribute__((ext_vector_type(8))) float v8f;

// ---------------------------------------------------------------------------
// Kernel 1: LayerNorm preprocess (unbiased std, ddof=1), save mu/std
// ---------------------------------------------------------------------------
__global__ __launch_bounds__(256) void preprocess_ln(
    const float* __restrict__ x, float* __restrict__ xn,
    float* __restrict__ muArr, float* __restrict__ stdArr)
{
    __shared__ float red[256];
    const int row = blockIdx.x;
    const int t   = threadIdx.x;
    const float* xr = x + (size_t)row * D_SZ;

    float v0 = xr[t];
    float v1 = xr[t + 256];

    red[t] = v0 + v1;
    __syncthreads();
    for (int s = 128; s > 0; s >>= 1) {
        if (t < s) red[t] += red[t + s];
        __syncthreads();
    }
    const float mu = red[0] * (1.0f / (float)D_SZ);
    __syncthreads();

    const float c0 = v0 - mu, c1 = v1 - mu;
    red[t] = c0 * c0 + c1 * c1;
    __syncthreads();
    for (int s = 128; s > 0; s >>= 1) {
        if (t < s) red[t] += red[t + s];
        __syncthreads();
    }
    const float var = red[0] * (1.0f / (float)(D_SZ - 1));
    const float sd  = sqrtf(var);
    const float inv = 1.0f / (sd + EPS_LN);

    xn[(size_t)row * D_SZ + t]       = c0 * inv;
    xn[(size_t)row * D_SZ + t + 256] = c1 * inv;
    if (t == 0) { muArr[row] = mu; stdArr[row] = sd; }
}

// ---------------------------------------------------------------------------
// Kernel 2: encode GEMM with fp32 WMMA (V_WMMA_F32_16X16X4_F32)
//   pre[b,l] = sum_d (xn[b,d]-pre_bias[d]) * W[l,d] + latent_bias[l]
//   Writes raw pre -> preRaw (latents_k slot, rewritten later by top-k)
//          pre*dead_mask -> preMasked output
// Block = 256 threads (8 waves). Each wave owns one 16x16 tile; a block covers
// a 16(M) x 128(N) output tile. A-tile staged in LDS (padded stride 516).
// ---------------------------------------------------------------------------
#define LDS_AS 516

__global__ __launch_bounds__(256) void encode_wmma(
    const float* __restrict__ xn, const float* __restrict__ Wm,
    const float* __restrict__ pre_bias, const float* __restrict__ latent_bias,
    const int* __restrict__ stats,
    float* __restrict__ preMasked, float* __restrict__ preRaw)
{
    __shared__ float As[16 * LDS_AS];
    const int t = threadIdx.x;
    const int mBase  = blockIdx.y * 16;
    const int nStrip = blockIdx.x * 128;

    // Cooperative, coalesced load of A tile = (xn - pre_bias), 16 x 512 f32
    #pragma unroll
    for (int i = 0; i < 8; ++i) {
        const int linear = i * 256 + t;       // float4 index within tile
        const int r  = linear >> 7;           // 128 float4 per row
        const int c4 = (linear & 127) << 2;
        const float4 xv = *(const float4*)(xn + (size_t)(mBase + r) * D_SZ + c4);
        const float4 bv = *(const float4*)(pre_bias + c4);
        float* dst = &As[r * LDS_AS + c4];
        dst[0] = xv.x - bv.x; dst[1] = xv.y - bv.y;
        dst[2] = xv.z - bv.z; dst[3] = xv.w - bv.w;
    }
    __syncthreads();

    const int lane = t & 31;
    const int wv   = t >> 5;
    const int hi   = lane >> 4;     // half-wave select
    const int nn   = lane & 15;     // M for A-frag, N for B-frag
    const int n    = nStrip + wv * 16 + nn;
    const float* wrow = Wm + (size_t)n * D_SZ;

    v8f c;
    #pragma unroll
    for (int i = 0; i < 8; ++i) c[i] = 0.0f;

    // K loop: 128 WMMA steps of K=4
    #pragma unroll 4
    for (int k0 = 0; k0 < D_SZ; k0 += 4) {
        const int ka = k0 + 2 * hi;
        // A fragment: lanes 0-15 hold (K=k0,k0+1) of row M; lanes 16-31 (K=k0+2,k0+3)
        const float2 av = *(const float2*)&As[nn * LDS_AS + ka];
        // B fragment: B[k][n] = W[n][k]  (consecutive k -> contiguous in W row)
        const float2 bv = *(const float2*)(wrow + ka);
        v2f a; a[0] = av.x; a[1] = av.y;
        v2f b; b[0] = bv.x; b[1] = bv.y;
        c = __builtin_amdgcn_wmma_f32_16x16x4_f32(
                /*neg_a=*/false, a, /*neg_b=*/false, b,
                /*c_mod=*/(short)0, c, /*reuse_a=*/false, /*reuse_b=*/false);
    }

    // Epilogue: + latent_bias, apply dead mask for pre_masked
    const float bias = latent_bias[n];
    const float dead = (stats[n] > DEAD_THRESHOLD) ? 1.0f : 0.0f;
    #pragma unroll
    for (int v = 0; v < 8; ++v) {
        const float p = c[v] + bias;
        const size_t off = (size_t)(mBase + v + 8 * hi) * (size_t)L_SZ + (size_t)n;
        preRaw[off]    = p;
        preMasked[off] = p * dead;
    }
}

// ---------------------------------------------------------------------------
// Kernel 3: exact per-row top-k via 4-pass radix select on ordered float keys,
// then deterministic ballot-based compaction. Optionally rewrites the row
// in-place as the scattered ReLU'd top-k (latents_k output).
// One block (256 threads = 8 waves) per row.
// ---------------------------------------------------------------------------
__device__ __forceinline__ unsigned fkey(float f) {
    unsigned b = __float_as_uint(f);
    return (b & 0x80000000u) ? ~b : (b | 0x80000000u);   // monotonic total order
}

__global__ __launch_bounds__(256) void topk_select(
    const float* in,            // [B, L] source (may alias rewrite)
    float* rewrite,             // nullptr, or [B, L] scatter+relu destination
    float* __restrict__ outVals, int* __restrict__ outIdx, int ksel)
{
    __shared__ unsigned sh_hist[256];
    __shared__ unsigned sh_bcast[2];
    __shared__ unsigned sh_weq[8], sh_wsel[8];
    __shared__ unsigned sh_eqBase, sh_selBase;

    const int row = blockIdx.x;
    const int t   = threadIdx.x;
    const float* src = in + (size_t)row * L_SZ;

    // ---- radix select: find threshold key T and tie quota ----
    unsigned prefix = 0;
    unsigned need   = (unsigned)ksel;
    for (int p = 3; p >= 0; --p) {
        sh_hist[t] = 0;
        __syncthreads();
        const unsigned maskHi = (p == 3) ? 0u : (0xFFFFFFFFu << (unsigned)(8 * (p + 1)));
        for (int i = t; i < L_SZ; i += 256) {
            const unsigned u = fkey(src[i]);
            if ((u & maskHi) == prefix)
                atomicAdd(&sh_hist[(u >> (8 * p)) & 0xFF], 1u);
        }
        __syncthreads();
        if (t == 0) {
            unsigned cum = 0, binSel = 0, newNeed = 1;
            int b2;
            for (b2 = 255; b2 >= 0; --b2) {
                const unsigned h = sh_hist[b2];
                if (cum + h >= need) { binSel = (unsigned)b2; newNeed = need - cum; break; }
                cum += h;
            }
            if (b2 < 0) { binSel = 0; newNeed = 1; }  // degenerate guard
            sh_bcast[0] = binSel; sh_bcast[1] = newNeed;
        }
        __syncthreads();
        prefix |= sh_bcast[0] << (8 * p);
        need    = sh_bcast[1];
    }
    const unsigned T     = prefix;
    const unsigned quota = need;   // number of ==T elements to accept (index order)

    if (t == 0) { sh_eqBase = 0; sh_selBase = 0; }
    __syncthreads();

    const int lane = t & 31;
    const int wid  = t >> 5;
    const unsigned laneLt = (lane == 0) ? 0u : (0xFFFFFFFFu >> (32 - lane));

    // ---- deterministic enumeration: exact ranks in index order ----
    for (int base = 0; base < L_SZ; base += 256) {
        const int i = base + t;
        const float v = src[i];
        const unsigned u = fkey(v);
        const bool eq = (u == T);

        const unsigned mEq = __builtin_amdgcn_ballot_w32(eq);
        if (lane == 0) sh_weq[wid] = (unsigned)__builtin_popcount(mEq);
        __syncthreads();
        unsigned eqLower = sh_eqBase;
        for (int w = 0; w < wid; ++w) eqLower += sh_weq[w];
        const unsigned eqRank = eqLower + (unsigned)__builtin_popcount(mEq & laneLt);

        const bool sel = (u > T) || (eq && (eqRank < quota));
        const unsigned mSel = __builtin_amdgcn_ballot_w32(sel);
        if (lane == 0) sh_wsel[wid] = (unsigned)__builtin_popcount(mSel);
        __syncthreads();
        unsigned pos = sh_selBase;
        for (int w = 0; w < wid; ++w) pos += sh_wsel[w];
        pos += (unsigned)__builtin_popcount(mSel & laneLt);

        if (sel && pos < (unsigned)ksel) {
            outVals[(size_t)row * ksel + pos] = fmaxf(v, 0.0f);
            outIdx [(size_t)row * ksel + pos] = i;
        }
        if (rewrite)
            rewrite[(size_t)row * L_SZ + i] = sel ? fmaxf(v, 0.0f) : 0.0f;

        __syncthreads();
        if (t == 0) {
            unsigned te = 0, ts = 0;
            for (int w = 0; w < 8; ++w) { te += sh_weq[w]; ts += sh_wsel[w]; }
            sh_eqBase += te; sh_selBase += ts;
        }
        __syncthreads();
    }
}

// ---------------------------------------------------------------------------
// Kernel 4: sparse decode + un-normalize
//   out[b,d] = (sum_k vals[k] * W[idx[k], d] + pre_bias[d]) * std[b] + mu[b]
// One block per row; sparse lists staged in LDS; W reads are L2-resident.
// ---------------------------------------------------------------------------
__global__ __launch_bounds__(256) void decode_sparse(
    const float* __restrict__ vals, const int* __restrict__ idx, int ksel,
    const float* __restrict__ Wm, const float* __restrict__ pre_bias,
    const float* __restrict__ muArr, const float* __restrict__ stdArr,
    float* __restrict__ out)
{
    __shared__ float sv[K_AUX];
    __shared__ int   si[K_AUX];
    const int row = blockIdx.x;
    const int t   = threadIdx.x;
    if (t < ksel) {
        sv[t] = vals[(size_t)row * ksel + t];
        si[t] = idx [(size_t)row * ksel + t];
    }
    __syncthreads();
    const float mu = muArr[row];
    const float sd = stdArr[row];
    for (int d = t; d < D_SZ; d += 256) {
        float acc = pre_bias[d];
        for (int k = 0; k < ksel; ++k)
            acc = fmaf(sv[k], Wm[(size_t)si[k] * D_SZ + d], acc);
        out[(size_t)row * D_SZ + d] = acc * sd + mu;
    }
}

// ---------------------------------------------------------------------------
// Launch
// ---------------------------------------------------------------------------
extern "C" void kernel_launch(void* const* d_in, const int* in_sizes, int n_in,
                              void* d_out, int out_size, void* d_ws, size_t ws_size,
                              hipStream_t stream)
{
    const float* x           = (const float*)d_in[0];   // [B, D]
    const float* Wm          = (const float*)d_in[1];   // [L, D]
    const float* pre_bias    = (const float*)d_in[2];   // [D]
    const float* latent_bias = (const float*)d_in[3];   // [L]
    const int*   stats       = (const int*)  d_in[4];   // [L]

    float* out = (float*)d_out;
    float* xn        = out;                                   // [B, D]
    float* preMasked = xn + (size_t)B_SZ * D_SZ;              // [B, L]
    float* latents   = preMasked + (size_t)B_SZ * L_SZ;       // [B, L]
    float* recons    = latents + (size_t)B_SZ * L_SZ;         // [B, D]
    float* reconsAux = recons + (size_t)B_SZ * D_SZ;          // [B, D]

    float* ws     = (float*)d_ws;
    float* muArr  = ws;
    float* stdArr = ws + B_SZ;
    float* tkV    = ws + 2 * B_SZ;
    int*   tkI    = (int*)(ws + 2 * B_SZ + (size_t)B_SZ * K_TOP);
    float* axV    = ws + 2 * B_SZ + 2 * (size_t)B_SZ * K_TOP;
    int*   axI    = (int*)(ws + 2 * B_SZ + 2 * (size_t)B_SZ * K_TOP + (size_t)B_SZ * K_AUX);

    // 1) LayerNorm preprocess -> xn, mu, std
    preprocess_ln<<<B_SZ, 256, 0, stream>>>(x, xn, muArr, stdArr);

    // 2) Encode GEMM (fp32 WMMA). Raw pre -> latents slot (temp), masked -> preMasked.
    encode_wmma<<<dim3(L_SZ / 128, B_SZ / 16), 256, 0, stream>>>(
        xn, Wm, pre_bias, latent_bias, stats, preMasked, latents);

    // 3) Top-K=32 on raw pre; rewrites latents slot as scattered ReLU'd top-k.
    topk_select<<<B_SZ, 256, 0, stream>>>(latents, latents, tkV, tkI, K_TOP);

    // 4) Aux Top-K=256 on masked pre (no rewrite; preMasked is a final output).
    topk_select<<<B_SZ, 256, 0, stream>>>(preMasked, nullptr, axV, axI, K_AUX);

    // 5) Sparse decodes + un-normalize.
    decode_sparse<<<B_SZ, 256, 0, stream>>>(tkV, tkI, K_TOP, Wm, pre_bias, muArr, stdArr, recons);
    decode_sparse<<<B_SZ, 256, 0, stream>>>(axV, axI, K_AUX, Wm, pre_bias, muArr, stdArr, reconsAux);
}